// TimestepNorm_16140487099030
// MI455X (gfx1250) — compile-verified
//
#include <hip/hip_runtime.h>
#include <stdint.h>

// Problem dims (fixed by the reference harness)
#define B_      8
#define L_      8192
#define D_      1024
#define S_      32           // number of L-segments for the Chan split
#define SEGLEN  256          // L_ / S_
#define TPB     256          // threads per block; 4 channels per thread => D_
#define VEC     4
#define PF      12           // outstanding async copies per wave
#define RING    16           // LDS ring slots (power of two, > PF)
#define EPSV    1e-5f
#define PRIORC  2.0f

// Native 16-byte vectors (HIP's float4 class is rejected by some builtins).
typedef float vf4 __attribute__((ext_vector_type(4)));
typedef int   vi4 __attribute__((vector_size(16)));

// gfx1250 async global->LDS path (b128 flavor), guarded so compilation never
// breaks. Probe-confirmed params: (AS1 int4*, AS3 int4*, imm offset, imm cpol).
#if defined(__AMDGCN__) && __has_builtin(__builtin_amdgcn_global_load_async_to_lds_b128) && __has_builtin(__builtin_amdgcn_s_wait_asynccnt)
#define TN_ASYNC 1
typedef vi4 __attribute__((address_space(1)))* tn_gp;
typedef vi4 __attribute__((address_space(3)))* tn_lp;
#else
#define TN_ASYNC 0
#endif

// ---------------------------------------------------------------------------
// Pass 1: per (b, segment, channel) local Welford stats (n, mean, M2).
// Grid: (S_, B_), block: TPB threads, 4 channels per thread (full D row).
// ---------------------------------------------------------------------------
__global__ __launch_bounds__(TPB) void tn_pass1(
    const float* __restrict__ x, const unsigned char* __restrict__ mask,
    float* __restrict__ seg_mean, float* __restrict__ seg_m2,
    float* __restrict__ seg_nvalid) {
  const int s  = blockIdx.x;
  const int b  = blockIdx.y;
  const int d0 = threadIdx.x * VEC;
  const int t0 = s * SEGLEN;
  const float* __restrict__ xp = x + ((size_t)b * L_ + t0) * D_ + d0;
  const unsigned char* __restrict__ mp = mask + (size_t)b * L_ + t0;

  float cnt = 0.0f;
  float m0 = 0.f, m1 = 0.f, m2 = 0.f, m3 = 0.f;
  float q0 = 0.f, q1 = 0.f, q2 = 0.f, q3 = 0.f;
#pragma unroll 4
  for (int t = 0; t < SEGLEN; ++t) {
    if (t + PF < SEGLEN) __builtin_prefetch(xp + (size_t)(t + PF) * D_, 0, 0);
    vf4 xv = *(const vf4*)(xp + (size_t)t * D_);
    if (!mp[t]) {
      cnt += 1.0f;
      float rc = __builtin_amdgcn_rcpf(cnt);
      float d;
      d = xv.x - m0; m0 += d * rc; q0 += d * (xv.x - m0);
      d = xv.y - m1; m1 += d * rc; q1 += d * (xv.y - m1);
      d = xv.z - m2; m2 += d * rc; q2 += d * (xv.z - m2);
      d = xv.w - m3; m3 += d * rc; q3 += d * (xv.w - m3);
    }
  }
  const size_t sidx = ((size_t)b * S_ + s) * D_ + d0;
  *(vf4*)(seg_mean + sidx) = (vf4){m0, m1, m2, m3};
  *(vf4*)(seg_m2   + sidx) = (vf4){q0, q1, q2, q3};
  if (threadIdx.x == 0) seg_nvalid[b * S_ + s] = cnt;
}

// ---------------------------------------------------------------------------
// Pass 2: sequential Chan-merge over the S_ segments per (b, d).
// Emits prefix carries for pass 3 and the final (count, mean, var) outputs.
// Grid: (B_), block: D_ threads.
// ---------------------------------------------------------------------------
__global__ __launch_bounds__(D_) void tn_pass2(
    const float* __restrict__ prior_mean, const float* __restrict__ prior_logv,
    const float* __restrict__ seg_mean, const float* __restrict__ seg_m2,
    const float* __restrict__ seg_nvalid,
    float* __restrict__ pre_mean, float* __restrict__ pre_m2,
    float* __restrict__ pre_count,
    float* __restrict__ out_count, float* __restrict__ out_mean,
    float* __restrict__ out_var) {
  const int d = threadIdx.x;
  const int b = blockIdx.x;

  float c  = PRIORC;
  float m  = prior_mean[d];
  float M2 = PRIORC * expf(prior_logv[d]);

  for (int s = 0; s < S_; ++s) {
    const size_t sidx = ((size_t)b * S_ + s) * D_ + d;
    pre_mean[sidx] = m;
    pre_m2[sidx]   = M2;
    if (d == 0) pre_count[b * S_ + s] = c;

    float nb = seg_nvalid[b * S_ + s];
    if (nb > 0.0f) {
      float mb  = seg_mean[sidx];
      float m2b = seg_m2[sidx];
      float cn  = c + nb;
      float r   = nb * __builtin_amdgcn_rcpf(cn);
      float delta = mb - m;
      m  += delta * r;
      M2 += m2b + delta * delta * c * r;
      c   = cn;
    }
  }
  out_mean[(size_t)b * D_ + d] = m;
  out_var [(size_t)b * D_ + d] = M2 * __builtin_amdgcn_rcpf(c);
  if (d == 0) out_count[b] = c;
}

// ---------------------------------------------------------------------------
// Pass 3: resume Welford from prefix carry; write y. x rows are staged through
// LDS with gfx1250 async global->LDS b128 copies (RING-slot ring, PF deep;
// each lane reads only its own 16B slot, so only s_wait_asynccnt gates reads.
// A slot is overwritten RING-PF iterations after its ds_load was consumed,
// so no explicit dscnt guard is needed).
// Grid: (S_, B_), block: TPB threads, 4 channels per thread.
// ---------------------------------------------------------------------------
__global__ __launch_bounds__(TPB) void tn_pass3(
    const float* __restrict__ x, const unsigned char* __restrict__ mask,
    const float* __restrict__ weight, const float* __restrict__ bias,
    const float* __restrict__ pre_mean, const float* __restrict__ pre_m2,
    const float* __restrict__ pre_count,
    float* __restrict__ y) {
  const int s  = blockIdx.x;
  const int b  = blockIdx.y;
  const int d0 = threadIdx.x * VEC;
  const int t0 = s * SEGLEN;
  const float* __restrict__ xp = x + ((size_t)b * L_ + t0) * D_ + d0;
  float*       __restrict__ yp = y + ((size_t)b * L_ + t0) * D_ + d0;
  const unsigned char* __restrict__ mp = mask + (size_t)b * L_ + t0;

  const size_t sidx = ((size_t)b * S_ + s) * D_ + d0;
  float c = pre_count[b * S_ + s];
  vf4 mv = *(const vf4*)(pre_mean + sidx);
  vf4 qv = *(const vf4*)(pre_m2   + sidx);
  float m0 = mv.x, m1 = mv.y, m2 = mv.z, m3 = mv.w;
  float q0 = qv.x, q1 = qv.y, q2 = qv.z, q3 = qv.w;
  vf4 wv = *(const vf4*)(weight + d0);
  vf4 bv = *(const vf4*)(bias   + d0);
  const float g0 = wv.x + 1.f, g1 = wv.y + 1.f, g2 = wv.z + 1.f, g3 = wv.w + 1.f;

#if TN_ASYNC
  __shared__ __align__(16) float stage[RING][D_];
#pragma unroll
  for (int k = 0; k < PF; ++k) {
    __builtin_amdgcn_global_load_async_to_lds_b128(
        (tn_gp)(uintptr_t)(xp + (size_t)k * D_),
        (tn_lp)(uintptr_t)&stage[k][d0], 0, 0);
  }
#endif

  for (int t = 0; t < SEGLEN; ++t) {
#if TN_ASYNC
    __builtin_amdgcn_s_wait_asynccnt(PF - 1);    // oldest copy (step t) done
    vf4 xv = *(const vf4*)&stage[t & (RING - 1)][d0];
#else
    if (t + PF < SEGLEN) __builtin_prefetch(xp + (size_t)(t + PF) * D_, 0, 0);
    vf4 xv = *(const vf4*)(xp + (size_t)t * D_);
#endif
    bool pad = (mp[t] != 0);
    if (!pad) {
      c += 1.0f;
      float rc = __builtin_amdgcn_rcpf(c);
      float d;
      d = xv.x - m0; m0 += d * rc; q0 += d * (xv.x - m0);
      d = xv.y - m1; m1 += d * rc; q1 += d * (xv.y - m1);
      d = xv.z - m2; m2 += d * rc; q2 += d * (xv.z - m2);
      d = xv.w - m3; m3 += d * rc; q3 += d * (xv.w - m3);
    }
    float rc2 = __builtin_amdgcn_rcpf(c);
    vf4 yv;
    yv.x = pad ? 0.f : (xv.x - m0) * __builtin_amdgcn_rsqf(q0 * rc2 + EPSV) * g0 + bv.x;
    yv.y = pad ? 0.f : (xv.y - m1) * __builtin_amdgcn_rsqf(q1 * rc2 + EPSV) * g1 + bv.y;
    yv.z = pad ? 0.f : (xv.z - m2) * __builtin_amdgcn_rsqf(q2 * rc2 + EPSV) * g2 + bv.z;
    yv.w = pad ? 0.f : (xv.w - m3) * __builtin_amdgcn_rsqf(q3 * rc2 + EPSV) * g3 + bv.w;
    __builtin_nontemporal_store(yv, (vf4*)(yp + (size_t)t * D_));

#if TN_ASYNC
    if (t + PF < SEGLEN) {
      __builtin_amdgcn_global_load_async_to_lds_b128(
          (tn_gp)(uintptr_t)(xp + (size_t)(t + PF) * D_),
          (tn_lp)(uintptr_t)&stage[(t + PF) & (RING - 1)][d0], 0, 0);
    }
#endif
  }
}

// ---------------------------------------------------------------------------
extern "C" void kernel_launch(void* const* d_in, const int* in_sizes, int n_in,
                              void* d_out, int out_size, void* d_ws, size_t ws_size,
                              hipStream_t stream) {
  const float*         x          = (const float*)d_in[0];
  const unsigned char* mask       = (const unsigned char*)d_in[1];
  const float*         prior_mean = (const float*)d_in[2];
  const float*         prior_logv = (const float*)d_in[3];
  const float*         weight     = (const float*)d_in[4];
  const float*         bias       = (const float*)d_in[5];

  float* out       = (float*)d_out;
  float* y         = out;                                   // [B,L,D]
  float* out_count = y + (size_t)B_ * L_ * D_;              // [B]
  float* out_mean  = out_count + B_;                        // [B,D]
  float* out_var   = out_mean + (size_t)B_ * D_;            // [B,D]

  float* ws = (float*)d_ws;
  const size_t SBD = (size_t)B_ * S_ * D_;
  float* seg_mean   = ws;                 // [B,S,D]
  float* seg_m2     = seg_mean + SBD;     // [B,S,D]
  float* pre_mean   = seg_m2 + SBD;       // [B,S,D]
  float* pre_m2     = pre_mean + SBD;     // [B,S,D]
  float* seg_nvalid = pre_m2 + SBD;       // [B,S]
  float* pre_count  = seg_nvalid + (size_t)B_ * S_;  // [B,S]

  dim3 g13(S_, B_);
  tn_pass1<<<g13, TPB, 0, stream>>>(x, mask, seg_mean, seg_m2, seg_nvalid);
  tn_pass2<<<dim3(B_), D_, 0, stream>>>(prior_mean, prior_logv,
                                        seg_mean, seg_m2, seg_nvalid,
                                        pre_mean, pre_m2, pre_count,
                                        out_count, out_mean, out_var);
  tn_pass3<<<g13, TPB, 0, stream>>>(x, mask, weight, bias,
                                    pre_mean, pre_m2, pre_count, y);
}